// EdgeSageLayer_9775345566348
// MI455X (gfx1250) — compile-verified
//
#include <hip/hip_runtime.h>
#include <hip/hip_bf16.h>

// ---------------------------------------------------------------------------
// EdgeSAGE layer for MI455X (gfx1250, wave32):
//   messages = concat(x[src], edge_attr) @ msg_w + msg_b      (E x 96 @ 96 x 64)
//   agg      = segment_mean(messages, dst)                    (scatter atomics)
//   out      = x @ self_w + self_b + agg                      (N x 64 @ 64 x 64)
// Both GEMMs use V_WMMA_F32_16X16X4_F32: one wave owns a 16x64 output tile.
// ---------------------------------------------------------------------------

typedef float v2f __attribute__((ext_vector_type(2)));
typedef float v8f __attribute__((ext_vector_type(8)));

__device__ __forceinline__ v8f wmma_f32_4(v2f a, v2f b, v8f c) {
    // D = A(16x4) * B(4x16) + C(16x16), fp32 accumulate
    return __builtin_amdgcn_wmma_f32_16x16x4_f32(
        /*neg_a=*/false, a, /*neg_b=*/false, b,
        /*c_mod=*/(short)0, c, /*reuse_a=*/false, /*reuse_b=*/false);
}

// Stage weight matrix W[K x 64] into LDS in per-lane WMMA B-fragment order:
// fragment (kb, nt, lane) = { W[4*kb + 2*hi    ][nt*16 + (lane&15)],
//                             W[4*kb + 2*hi + 1][nt*16 + (lane&15)] },  hi = lane>>4
// so the inner loop is a single contiguous ds_load_b64 per (kb, nt).
template <int KB>
__device__ __forceinline__ void stage_B(const float* __restrict__ w, float* __restrict__ swB,
                                        int tid, int nthreads) {
    for (int i = tid; i < KB * 4 * 32; i += nthreads) {
        int lane = i & 31;
        int nt   = (i >> 5) & 3;
        int kb   = i >> 7;
        int hi   = lane >> 4;
        int k0   = kb * 4 + 2 * hi;
        int n    = nt * 16 + (lane & 15);
        swB[2 * i + 0] = w[(k0 + 0) * 64 + n];
        swB[2 * i + 1] = w[(k0 + 1) * 64 + n];
    }
}

// ---------------------------------------------------------------------------
// Kernel 1: per 16-edge tile, compute 16x64 message tile with WMMA (K = 96 =
// 24 k-blocks of 4), then scatter-add rows to agg[dst] and bump deg[dst].
// ---------------------------------------------------------------------------
__global__ __launch_bounds__(256) void edge_msg_kernel(
    const float* __restrict__ x,
    const long long* __restrict__ srcp,
    const long long* __restrict__ dstp,
    const float* __restrict__ edge_attr,
    const float* __restrict__ msg_w,
    const float* __restrict__ msg_b,
    float* __restrict__ agg,
    float* __restrict__ deg,
    int E, int numTiles)
{
    __shared__ float swB[24 * 4 * 32 * 2];   // 24 KB, pre-swizzled msg_w
    stage_B<24>(msg_w, swB, threadIdx.x, blockDim.x);
    __syncthreads();

    const int lane = threadIdx.x & 31;
    const int wave = threadIdx.x >> 5;
    const int tile = blockIdx.x * (blockDim.x >> 5) + wave;
    if (tile >= numTiles) return;

    const int hi = lane >> 4;        // half-wave selects K sub-pair / row group
    const int m  = lane & 15;        // edge-in-tile for A loads, out column for C

    // Edge for this lane's A fragments (clamped for tail safety; WMMA needs
    // EXEC all-ones, so we compute garbage rows and simply don't scatter them).
    long long e      = (long long)tile * 16 + m;
    long long eclamp = (e < (long long)E) ? e : (long long)(E - 1);
    const int src_e  = (int)srcp[eclamp];
    const float* xrow = x + (size_t)src_e * 64;
    const float* arow = edge_attr + (size_t)eclamp * 32;

    // Degree scatter (one lane per edge).
    if (hi == 0 && e < (long long)E) {
        atomicAdd(deg + (int)dstp[e], 1.0f);
    }

    // Accumulators seeded with bias (D element (M,n) gets + msg_b[n]).
    const float b0 = msg_b[ 0 + m], b1 = msg_b[16 + m],
                b2 = msg_b[32 + m], b3 = msg_b[48 + m];
    v8f c0 = {b0, b0, b0, b0, b0, b0, b0, b0};
    v8f c1 = {b1, b1, b1, b1, b1, b1, b1, b1};
    v8f c2 = {b2, b2, b2, b2, b2, b2, b2, b2};
    v8f c3 = {b3, b3, b3, b3, b3, b3, b3, b3};

    const v2f* bf = (const v2f*)swB;
    #pragma unroll
    for (int kb = 0; kb < 24; ++kb) {
        v2f a;
        if (kb < 16) a = *(const v2f*)(xrow + kb * 4 + 2 * hi);          // x[src] part
        else         a = *(const v2f*)(arow + (kb - 16) * 4 + 2 * hi);   // edge_attr part
        v2f bb0 = bf[(kb * 4 + 0) * 32 + lane];
        v2f bb1 = bf[(kb * 4 + 1) * 32 + lane];
        v2f bb2 = bf[(kb * 4 + 2) * 32 + lane];
        v2f bb3 = bf[(kb * 4 + 3) * 32 + lane];
        c0 = wmma_f32_4(a, bb0, c0);
        c1 = wmma_f32_4(a, bb1, c1);
        c2 = wmma_f32_4(a, bb2, c2);
        c3 = wmma_f32_4(a, bb3, c3);
    }

    // Scatter-add: C layout => lane holds rows M = r + 8*hi, column m of each n-tile.
    #pragma unroll
    for (int r = 0; r < 8; ++r) {
        long long er = (long long)tile * 16 + r + 8 * hi;
        if (er < (long long)E) {
            int d = (int)dstp[er];
            float* basep = agg + (size_t)d * 64 + m;
            atomicAdd(basep +  0, c0[r]);
            atomicAdd(basep + 16, c1[r]);
            atomicAdd(basep + 32, c2[r]);
            atomicAdd(basep + 48, c3[r]);
        }
    }
}

// ---------------------------------------------------------------------------
// Kernel 2: out = x @ self_w + self_b + agg / max(deg, 1).  16-node tiles,
// K = 64 = 16 k-blocks of 4.
// ---------------------------------------------------------------------------
__global__ __launch_bounds__(256) void node_out_kernel(
    const float* __restrict__ x,
    const float* __restrict__ self_w,
    const float* __restrict__ self_b,
    const float* __restrict__ agg,
    const float* __restrict__ deg,
    float* __restrict__ out,
    int N, int numTiles)
{
    __shared__ float swB[16 * 4 * 32 * 2];   // 16 KB, pre-swizzled self_w
    stage_B<16>(self_w, swB, threadIdx.x, blockDim.x);
    __syncthreads();

    const int lane = threadIdx.x & 31;
    const int wave = threadIdx.x >> 5;
    const int tile = blockIdx.x * (blockDim.x >> 5) + wave;
    if (tile >= numTiles) return;

    const int hi = lane >> 4;
    const int m  = lane & 15;

    long long row    = (long long)tile * 16 + m;
    long long rclamp = (row < (long long)N) ? row : (long long)(N - 1);
    const float* xrow = x + (size_t)rclamp * 64;

    const float b0 = self_b[ 0 + m], b1 = self_b[16 + m],
                b2 = self_b[32 + m], b3 = self_b[48 + m];
    v8f c0 = {b0, b0, b0, b0, b0, b0, b0, b0};
    v8f c1 = {b1, b1, b1, b1, b1, b1, b1, b1};
    v8f c2 = {b2, b2, b2, b2, b2, b2, b2, b2};
    v8f c3 = {b3, b3, b3, b3, b3, b3, b3, b3};

    const v2f* bf = (const v2f*)swB;
    #pragma unroll
    for (int kb = 0; kb < 16; ++kb) {
        v2f a = *(const v2f*)(xrow + kb * 4 + 2 * hi);
        v2f bb0 = bf[(kb * 4 + 0) * 32 + lane];
        v2f bb1 = bf[(kb * 4 + 1) * 32 + lane];
        v2f bb2 = bf[(kb * 4 + 2) * 32 + lane];
        v2f bb3 = bf[(kb * 4 + 3) * 32 + lane];
        c0 = wmma_f32_4(a, bb0, c0);
        c1 = wmma_f32_4(a, bb1, c1);
        c2 = wmma_f32_4(a, bb2, c2);
        c3 = wmma_f32_4(a, bb3, c3);
    }

    // Finalize: add mean-aggregated messages and store.
    #pragma unroll
    for (int r = 0; r < 8; ++r) {
        long long rr = (long long)tile * 16 + r + 8 * hi;
        if (rr < (long long)N) {
            float dg  = deg[rr];
            float inv = 1.0f / (dg > 1.0f ? dg : 1.0f);
            const float* arow = agg + (size_t)rr * 64 + m;
            float* orow       = out + (size_t)rr * 64 + m;
            orow[ 0] = c0[r] + arow[ 0] * inv;
            orow[16] = c1[r] + arow[16] * inv;
            orow[32] = c2[r] + arow[32] * inv;
            orow[48] = c3[r] + arow[48] * inv;
        }
    }
}

// ---------------------------------------------------------------------------
// Inputs (setup_inputs order): x[N,64] f32, edge_index[2,E] i64,
// edge_attr[E,32] f32, msg_w[96,64] f32, msg_b[64] f32, self_w[64,64] f32,
// self_b[64] f32.  Output: [N,64] f32.  Workspace: agg[N*64] + deg[N] f32.
// ---------------------------------------------------------------------------
extern "C" void kernel_launch(void* const* d_in, const int* in_sizes, int n_in,
                              void* d_out, int out_size, void* d_ws, size_t ws_size,
                              hipStream_t stream) {
    const float*     x          = (const float*)d_in[0];
    const long long* edge_index = (const long long*)d_in[1];
    const float*     edge_attr  = (const float*)d_in[2];
    const float*     msg_w      = (const float*)d_in[3];
    const float*     msg_b      = (const float*)d_in[4];
    const float*     self_w     = (const float*)d_in[5];
    const float*     self_b     = (const float*)d_in[6];
    float*           out        = (float*)d_out;

    const int N = in_sizes[0] / 64;            // nodes
    const int E = in_sizes[2] / 32;            // edges
    const long long* srcp = edge_index;        // edge_index[0]
    const long long* dstp = edge_index + E;    // edge_index[1]

    float* agg = (float*)d_ws;                 // [N, 64]
    float* deg = agg + (size_t)N * 64;         // [N]
    hipMemsetAsync(d_ws, 0, ((size_t)N * 64 + N) * sizeof(float), stream);

    const int tilesE  = (E + 15) / 16;         // one wave32 per 16-edge tile
    const int blocksE = (tilesE + 7) / 8;      // 8 waves / 256-thread block
    edge_msg_kernel<<<blocksE, 256, 0, stream>>>(x, srcp, dstp, edge_attr,
                                                 msg_w, msg_b, agg, deg, E, tilesE);

    const int tilesN  = (N + 15) / 16;         // one wave32 per 16-node tile
    const int blocksN = (tilesN + 7) / 8;
    node_out_kernel<<<blocksN, 256, 0, stream>>>(x, self_w, self_b, agg, deg,
                                                 out, N, tilesN);
}